// MultiboxLoss_49039936586274
// MI455X (gfx1250) — compile-verified
//
#include <hip/hip_runtime.h>

// MultiboxLoss for MI455X (gfx1250, wave32).
// Inputs (setup_inputs order):
//   d_in[0] classes      f32 [B, A, C]   (log-softmax scores)
//   d_in[1] locs         f32 [B, A, 4]
//   d_in[2] target_locs  f32 [B, A, 4]
//   d_in[3] target_cls   int [B, A]
// Output: 3 floats (loss, cls_loss, loc_loss)

#define AA  8732
#define CC  81
#define TPB 1024                      // 32 wave32 waves, one workgroup per batch row
#define VPT ((AA + TPB - 1) / TPB)    // 9 anchors per thread, register-resident

typedef __attribute__((ext_vector_type(2))) float v2f;
typedef __attribute__((ext_vector_type(8))) float v8f;

__device__ __forceinline__ float smooth_l1(float x) {
    float ax = fabsf(x);
    return ax < 1.0f ? 0.5f * x * x : ax - 0.5f;
}

// Fused block reduction of (int, float, float): wave32 shuffle chains + one
// 32-slot LDS hop. Two barriers.
__device__ __forceinline__ void blk_red_iff(int& pi, float& f1, float& f2,
                                            int* s_i, float* s_f1, float* s_f2,
                                            int tid) {
    const int wave = tid >> 5, lane = tid & 31;
    #pragma unroll
    for (int off = 16; off > 0; off >>= 1) {
        pi += __shfl_down(pi, off, 32);
        f1 += __shfl_down(f1, off, 32);
        f2 += __shfl_down(f2, off, 32);
    }
    if (lane == 0) { s_i[wave] = pi; s_f1[wave] = f1; s_f2[wave] = f2; }
    __syncthreads();
    if (wave == 0) {
        int a = s_i[lane]; float b = s_f1[lane]; float c = s_f2[lane];
        #pragma unroll
        for (int off = 16; off > 0; off >>= 1) {
            a += __shfl_down(a, off, 32);
            b += __shfl_down(b, off, 32);
            c += __shfl_down(c, off, 32);
        }
        if (lane == 0) { s_i[0] = a; s_f1[0] = b; s_f2[0] = c; }
    }
    __syncthreads();
    pi = s_i[0]; f1 = s_f1[0]; f2 = s_f2[0];
}

// Single-float block reduction (same structure).
__device__ __forceinline__ float blk_red_f(float v, float* s, int tid) {
    const int wave = tid >> 5, lane = tid & 31;
    #pragma unroll
    for (int off = 16; off > 0; off >>= 1) v += __shfl_down(v, off, 32);
    if (lane == 0) s[wave] = v;
    __syncthreads();
    if (wave == 0) {
        float x = s[lane];
        #pragma unroll
        for (int off = 16; off > 0; off >>= 1) x += __shfl_down(x, off, 32);
        if (lane == 0) s[0] = x;
    }
    __syncthreads();
    return s[0];
}

// Block reduction of an 8-uint packed histogram (16 bins, 2x16-bit per uint).
// Staging slots [0,256), result slots [256,264) -> only 2 barriers per call,
// safe for back-to-back reuse across radix passes.
__device__ __forceinline__ void blk_red_hist8(unsigned h[8], unsigned* sbuf, int tid) {
    const int wave = tid >> 5, lane = tid & 31;
    #pragma unroll
    for (int i = 0; i < 8; ++i) {
        unsigned v = h[i];
        #pragma unroll
        for (int off = 16; off > 0; off >>= 1) v += __shfl_down(v, off, 32);
        h[i] = v;
    }
    if (lane == 0) {
        #pragma unroll
        for (int i = 0; i < 8; ++i) sbuf[wave * 8 + i] = h[i];
    }
    __syncthreads();
    if (wave == 0) {
        unsigned t[8];
        #pragma unroll
        for (int i = 0; i < 8; ++i) t[i] = sbuf[lane * 8 + i];
        #pragma unroll
        for (int i = 0; i < 8; ++i) {
            unsigned v = t[i];
            #pragma unroll
            for (int off = 16; off > 0; off >>= 1) v += __shfl_down(v, off, 32);
            t[i] = v;
        }
        if (lane == 0) {
            #pragma unroll
            for (int i = 0; i < 8; ++i) sbuf[256 + i] = t[i];
        }
    }
    __syncthreads();
    #pragma unroll
    for (int i = 0; i < 8; ++i) h[i] = sbuf[256 + i];
}

__global__ void mbox_init_kernel(float* ws_f, int* ws_i) {
    ws_f[0] = 0.0f;   // global cls sum
    ws_f[1] = 0.0f;   // global loc sum
    ws_i[2] = 0;      // global positive count
}

__global__ __launch_bounds__(TPB)
void mbox_row_kernel(const float* __restrict__ cls,
                     const float* __restrict__ locs,
                     const float* __restrict__ tlocs,
                     const int*   __restrict__ tcls,
                     float* ws_f, int* ws_i) {
    __shared__ unsigned s_h[264];     // histogram staging (256) + result (8)
    __shared__ int      s_ri[32];
    __shared__ float    s_rf1[32];
    __shared__ float    s_rf2[32];

    const int b   = blockIdx.x;
    const int tid = threadIdx.x;

    const int*    trow  = tcls  + (size_t)b * AA;
    const float*  crow  = cls   + (size_t)b * AA * CC;
    const float4* lrow  = (const float4*)locs  + (size_t)b * AA;
    const float4* tlrow = (const float4*)tlocs + (size_t)b * AA;

    // ---- Phase 1: gather class NLL, pos/neg bookkeeping, smooth-L1 on positives.
    // Negative losses (>= 0) stay register-resident; -1 sentinel elsewhere.
    float val[VPT];
    int   pos_c = 0, neg_c = 0, cand = 0;
    float cls_pos = 0.0f, loc_s = 0.0f;
    #pragma unroll
    for (int j = 0; j < VPT; ++j) {
        int a = tid + j * TPB;
        float v = -1.0f;
        if (a < AA) {
            if (a + TPB < AA)
                __builtin_prefetch(trow + a + TPB, 0, 0);   // global_prefetch_b8
            int  tc    = trow[a];
            bool valid = (tc != -1);
            int  idx   = tc > 0 ? tc : 0;
            float cl   = valid ? -crow[(size_t)a * CC + idx] : 0.0f;  // NLL >= 0
            bool pos = tc > 0, neg = tc == 0;
            if (neg) { v = cl; neg_c++; cand |= (1 << j); }
            if (pos) {
                pos_c++;
                cls_pos += cl;
                float4 l = lrow[a], t = tlrow[a];
                loc_s += smooth_l1(l.x - t.x) + smooth_l1(l.y - t.y)
                       + smooth_l1(l.z - t.z) + smooth_l1(l.w - t.w);
            }
        }
        val[j] = v;
    }

    // Fused reduction: pos|neg packed 16/16 (each <= 8732 < 2^16, carry-free).
    int   pn = pos_c | (neg_c << 16);
    float cls_pos_t = cls_pos, loc_t = loc_s;
    blk_red_iff(pn, cls_pos_t, loc_t, s_ri, s_rf1, s_rf2, tid);
    const int num_pos = pn & 0xFFFF;
    const int num_neg = (pn >> 16) & 0xFFFF;
    const int k       = min(3 * num_pos, num_neg);   // NEG_PER_POS_RATIO=3, MIN_NEG=0

    // ---- Phase 2: k-th largest negative loss via MSB-first 4-bit radix select.
    // Real losses are non-negative floats -> bit pattern is order-preserving.
    // `cand` tracks which register values still match the selected prefix, so no
    // prefix compare is needed. Loop NOT unrolled (one body copy, small I$
    // footprint); per-value loops unrolled so val[]/bits stay in registers.
    float topk = 0.0f;
    if (k > 0) {                      // uniform per block
        unsigned prefix = 0u;
        int kp = k;                   // rank within current candidate set
        #pragma unroll 1
        for (int shift = 28; shift >= 0; shift -= 4) {
            unsigned h[8] = {0u, 0u, 0u, 0u, 0u, 0u, 0u, 0u};
            #pragma unroll
            for (int j = 0; j < VPT; ++j) {
                if (cand & (1 << j)) {
                    unsigned d = (__float_as_uint(val[j]) >> shift) & 15u;
                    h[d >> 1] += 1u << ((d & 1u) * 16u);
                }
            }
            blk_red_hist8(h, s_h, tid);
            // Deterministic digit pick, identical on every thread (no broadcast).
            int cum = 0;
            unsigned dsel = 0u;
            #pragma unroll
            for (int e = 15; e >= 0; --e) {
                int cnt = (int)((h[e >> 1] >> ((e & 1) * 16)) & 0xFFFFu);
                if (cum + cnt >= kp) { dsel = (unsigned)e; kp -= cum; break; }
                cum += cnt;
            }
            prefix = (prefix << 4) | dsel;
            // Prune candidates whose nibble != selected digit.
            #pragma unroll
            for (int j = 0; j < VPT; ++j) {
                if (((__float_as_uint(val[j]) >> shift) & 15u) != dsel)
                    cand &= ~(1 << j);
            }
        }
        const float T = __uint_as_float(prefix);   // k-th largest value (exists)
        // Every v > T was counted in exactly one pass's `cum`, so
        // count(v > T) = k - kp and (k - cnt_gt) = kp. No extra reduction needed.

        // ---- Phase 3: sum of strictly-greater values on the matrix pipe.
        // D = A(16x4) x ones(4x16) + C : every column of D holds the row-sums, so
        // sum(all 256 D entries) = 16 * sum(inputs). Fully unrolled, branch-free
        // -> EXEC is all-ones at every WMMA as the ISA requires.
        v8f acc = {0.f, 0.f, 0.f, 0.f, 0.f, 0.f, 0.f, 0.f};
        v2f ones; ones.x = 1.0f; ones.y = 1.0f;
        #pragma unroll
        for (int j = 0; j < VPT; j += 2) {
            float v0 = val[j];
            float v1 = (j + 1 < VPT) ? val[j + 1] : -1.0f;
            v2f aop;
            aop.x = (v0 > T) ? v0 : 0.0f;   // T >= 0, sentinels never pass
            aop.y = (v1 > T) ? v1 : 0.0f;
            acc = __builtin_amdgcn_wmma_f32_16x16x4_f32(
                false, aop, false, ones, (short)0, acc, false, false);
        }
        float ps = acc[0] + acc[1] + acc[2] + acc[3] + acc[4] + acc[5] + acc[6] + acc[7];
        float sum16 = blk_red_f(ps, s_rf1, tid);        // 16x the true masked sum
        topk = sum16 * 0.0625f + (float)kp * T;         // ties at T contribute T each
    }

    if (tid == 0) {
        atomicAdd(&ws_f[0], cls_pos_t + topk);
        atomicAdd(&ws_f[1], loc_t);
        atomicAdd(&ws_i[2], num_pos);
    }
}

__global__ void mbox_final_kernel(const float* ws_f, const int* ws_i, float* out) {
    int   np  = ws_i[2];
    float div = (float)(np > 1 ? np : 1);
    float c = ws_f[0] / div;
    float l = ws_f[1] / div;
    out[0] = c + l;   // loss (CLS_WEIGHT = LOC_WEIGHT = 1)
    out[1] = c;       // class loss
    out[2] = l;       // loc loss
}

extern "C" void kernel_launch(void* const* d_in, const int* in_sizes, int n_in,
                              void* d_out, int out_size, void* d_ws, size_t ws_size,
                              hipStream_t stream) {
    const float* cls   = (const float*)d_in[0];
    const float* locs  = (const float*)d_in[1];
    const float* tlocs = (const float*)d_in[2];
    const int*   tcls  = (const int*)d_in[3];
    float* out  = (float*)d_out;
    float* ws_f = (float*)d_ws;
    int*   ws_i = (int*)d_ws;

    const int B = in_sizes[3] / AA;   // 64

    mbox_init_kernel<<<1, 1, 0, stream>>>(ws_f, ws_i);
    mbox_row_kernel<<<B, TPB, 0, stream>>>(cls, locs, tlocs, tcls, ws_f, ws_i);
    mbox_final_kernel<<<1, 1, 0, stream>>>(ws_f, ws_i, out);
}